// SinkhornProjection_3315714753047
// MI455X (gfx1250) — compile-verified
//
#include <hip/hip_runtime.h>
#include <hip/hip_bf16.h>

// Sinkhorn on 8192 x (64x64) f32.  u/v scaling-vector formulation:
//   E = exp(x);  u <- u/(u*(E v)+EPS);  v <- v/(v*(E^T u)+EPS);  out = diag(u) E diag(v)
// 2 waves per matrix: wave0 holds E in WMMA A-layout (computes E*v via
// v_wmma_f32_16x16x4_f32 with broadcast-B), wave1 holds E in B-layout
// (computes E^T*u).  E staged once via async global->LDS DMA.
// The 4 accumulator chains are interleaved (K outer, tile inner) so the
// matrix pipe sees 4 independent WMMA dependency chains.

#define ITERS     20
#define SINK_EPS  1e-9f
#define LDSTRIDE  68   // padded row stride (floats) in LDS stage

typedef float v2f __attribute__((ext_vector_type(2)));
typedef float v4f __attribute__((ext_vector_type(4)));
typedef float v8f __attribute__((ext_vector_type(8)));
typedef int   v4i __attribute__((ext_vector_type(4)));

typedef __attribute__((address_space(1))) v4i* gv4i_p;   // global int4*
typedef __attribute__((address_space(3))) v4i* lv4i_p;   // LDS int4*

#if __has_builtin(__builtin_amdgcn_wmma_f32_16x16x4_f32)
#define HAVE_WMMA_F32 1
#else
#define HAVE_WMMA_F32 0
#endif

#if __has_builtin(__builtin_amdgcn_global_load_async_to_lds_b128)
#define HAVE_ASYNC 1
#else
#define HAVE_ASYNC 0
#endif

#if HAVE_WMMA_F32
__device__ __forceinline__ v8f wmma16x16x4(v2f a, v2f b, v8f c) {
    // 8 args: (neg_a, A, neg_b, B, c_mod, C, reuse_a, reuse_b)
    return __builtin_amdgcn_wmma_f32_16x16x4_f32(false, a, false, b, (short)0, c,
                                                 false, false);
}
#endif

__global__ void __launch_bounds__(64)
sinkhorn_uv_kernel(const float* __restrict__ X, float* __restrict__ Out)
{
    __shared__ float eStage[64 * LDSTRIDE];   // padded 64x64 stage (~17.4 KB)
    __shared__ float uLDS[64];
    __shared__ float vLDS[64];
#if HAVE_WMMA_F32
    __shared__ float sLDS[64];
#endif

    const int tid  = threadIdx.x;       // 0..63
    const int wave = tid >> 5;          // 0 or 1
    const int lane = tid & 31;
    const int half = lane >> 4;         // 0: lanes 0-15, 1: lanes 16-31
    const int l16  = lane & 15;

    const size_t base = (size_t)blockIdx.x * 4096u;
    const float* xm = X + base;
    float*       om = Out + base;

    // ---- Stage raw x into LDS: 1024 16B chunks across 64 threads ----
#if HAVE_ASYNC
    #pragma unroll
    for (int i = 0; i < 16; ++i) {
        const int t = i * 64 + tid;
        const int m = t >> 4, j = t & 15;
        __builtin_amdgcn_global_load_async_to_lds_b128(
            (gv4i_p)(xm + m * 64 + j * 4),
            (lv4i_p)(&eStage[m * LDSTRIDE + j * 4]),
            0, 0);
    }
  #if __has_builtin(__builtin_amdgcn_s_wait_asynccnt)
    __builtin_amdgcn_s_wait_asynccnt(0);
  #else
    asm volatile("s_wait_asynccnt 0" ::: "memory");
  #endif
#else
    #pragma unroll
    for (int i = 0; i < 16; ++i) {
        const int t = i * 64 + tid;
        const int m = t >> 4, j = t & 15;
        *(v4f*)(&eStage[m * LDSTRIDE + j * 4]) = *(const v4f*)(xm + m * 64 + j * 4);
    }
#endif
    vLDS[tid] = 1.0f;
    __syncthreads();

    // ---- Gather E = exp(x) into per-wave register layout (128 VGPRs) ----
    // wave0: Em[tr*16+k] = A-layout pair: rows 16tr+l16, cols 4k+2*half,+1
    // wave1: Em[q*4+tc]  = B-layout pair: rows 4q+2*half,+1, col 16tc+l16
    v2f Em[64];
    float g[4];                         // wave0: u rows; wave1: v cols
    #pragma unroll
    for (int i = 0; i < 4; ++i) g[i] = 1.0f;

    if (wave == 0) {
        #pragma unroll
        for (int tr = 0; tr < 4; ++tr) {
            const int m = tr * 16 + l16;
            #pragma unroll
            for (int k = 0; k < 16; ++k) {
                v2f raw = *(const v2f*)(&eStage[m * LDSTRIDE + k * 4 + 2 * half]);
                v2f e; e[0] = __expf(raw[0]); e[1] = __expf(raw[1]);
                Em[tr * 16 + k] = e;
            }
        }
    } else {
        #pragma unroll
        for (int q = 0; q < 16; ++q) {
            const int r0 = q * 4 + 2 * half;
            #pragma unroll
            for (int tc = 0; tc < 4; ++tc) {
                const int c = tc * 16 + l16;
                v2f e;
                e[0] = __expf(eStage[r0 * LDSTRIDE + c]);
                e[1] = __expf(eStage[(r0 + 1) * LDSTRIDE + c]);
                Em[q * 4 + tc] = e;
            }
        }
    }

    // ---- 20 Sinkhorn iterations on u/v only ----
    for (int it = 0; it < ITERS; ++it) {
        __syncthreads();                     // vLDS ready
        if (wave == 0) {
            v2f bv[16];                      // B operand: v broadcast per K-chunk
            #pragma unroll
            for (int k = 0; k < 16; ++k)
                bv[k] = *(const v2f*)(&vLDS[k * 4 + 2 * half]);
#if HAVE_WMMA_F32
            v8f acc[4] = {{}, {}, {}, {}};   // 4 independent WMMA chains
            #pragma unroll
            for (int k = 0; k < 16; ++k) {
                #pragma unroll
                for (int tr = 0; tr < 4; ++tr)
                    acc[tr] = wmma16x16x4(Em[tr * 16 + k], bv[k], acc[tr]);
            }
            // acc[tr][r] = (E v)[16tr + r + 8*half], replicated over 16 lanes
            if (l16 == 0) {
                #pragma unroll
                for (int tr = 0; tr < 4; ++tr) {
                    #pragma unroll
                    for (int r = 0; r < 8; ++r)
                        sLDS[tr * 16 + half * 8 + r] = acc[tr][r];
                }
            }
            #pragma unroll
            for (int tr = 0; tr < 4; ++tr) {
                const float s = sLDS[tr * 16 + l16];
                g[tr] = g[tr] * __builtin_amdgcn_rcpf(fmaf(g[tr], s, SINK_EPS));
                if (half == 0) uLDS[tr * 16 + l16] = g[tr];
            }
#else
            #pragma unroll
            for (int tr = 0; tr < 4; ++tr) {
                float p = 0.0f;
                #pragma unroll
                for (int k = 0; k < 16; ++k) {
                    v2f e = Em[tr * 16 + k];
                    p = fmaf(e[0], bv[k][0], p);
                    p = fmaf(e[1], bv[k][1], p);
                }
                const float s = p + __shfl_xor(p, 16, 32);
                g[tr] = g[tr] * __builtin_amdgcn_rcpf(fmaf(g[tr], s, SINK_EPS));
                if (half == 0) uLDS[tr * 16 + l16] = g[tr];
            }
#endif
        }
        __syncthreads();                     // uLDS ready
        if (wave == 1) {
            v2f au[16];                      // A operand: u broadcast per K-chunk
            #pragma unroll
            for (int q = 0; q < 16; ++q)
                au[q] = *(const v2f*)(&uLDS[q * 4 + 2 * half]);
#if HAVE_WMMA_F32
            v8f accv[4] = {{}, {}, {}, {}};  // 4 independent WMMA chains
            #pragma unroll
            for (int q = 0; q < 16; ++q) {
                #pragma unroll
                for (int tc = 0; tc < 4; ++tc)
                    accv[tc] = wmma16x16x4(au[q], Em[q * 4 + tc], accv[tc]);
            }
            #pragma unroll
            for (int tc = 0; tc < 4; ++tc) {
                const float w = accv[tc][0]; // (E^T u)[16tc + l16], all rows equal
                g[tc] = g[tc] * __builtin_amdgcn_rcpf(fmaf(g[tc], w, SINK_EPS));
                if (half == 0) vLDS[tc * 16 + l16] = g[tc];
            }
#else
            #pragma unroll
            for (int tc = 0; tc < 4; ++tc) {
                float p = 0.0f;
                #pragma unroll
                for (int q = 0; q < 16; ++q) {
                    v2f e = Em[q * 4 + tc];
                    p = fmaf(au[q][0], e[0], p);
                    p = fmaf(au[q][1], e[1], p);
                }
                const float w = p + __shfl_xor(p, 16, 32);
                g[tc] = g[tc] * __builtin_amdgcn_rcpf(fmaf(g[tc], w, SINK_EPS));
                if (half == 0) vLDS[tc * 16 + l16] = g[tc];
            }
#endif
        }
    }
    __syncthreads();                         // final u (uLDS) / v (vLDS) published

    // ---- out = diag(u) E diag(v); wave0 writes rows 0-31, wave1 rows 32-63 ----
    if (wave == 0) {
        #pragma unroll
        for (int tr = 0; tr < 2; ++tr) {
            const int m = tr * 16 + l16;
            #pragma unroll
            for (int k = 0; k < 16; ++k) {
                v2f vv = *(const v2f*)(&vLDS[k * 4 + 2 * half]);
                v2f e  = Em[tr * 16 + k];
                v2f o;
                o[0] = e[0] * g[tr] * vv[0];
                o[1] = e[1] * g[tr] * vv[1];
                *(v2f*)(om + m * 64 + k * 4 + 2 * half) = o;
            }
        }
    } else {
        #pragma unroll
        for (int q = 8; q < 16; ++q) {
            v2f uu = *(const v2f*)(&uLDS[q * 4 + 2 * half]);
            const int r0 = q * 4 + 2 * half;
            #pragma unroll
            for (int tc = 0; tc < 4; ++tc) {
                const int c = tc * 16 + l16;
                v2f e = Em[q * 4 + tc];
                om[r0 * 64 + c]       = e[0] * uu[0] * g[tc];
                om[(r0 + 1) * 64 + c] = e[1] * uu[1] * g[tc];
            }
        }
    }
}

extern "C" void kernel_launch(void* const* d_in, const int* in_sizes, int n_in,
                              void* d_out, int out_size, void* d_ws, size_t ws_size,
                              hipStream_t stream) {
    (void)n_in; (void)out_size; (void)d_ws; (void)ws_size;
    const float* x = (const float*)d_in[0];
    float* out = (float*)d_out;
    const int nmat = in_sizes[0] / 4096;     // 8192 matrices of 64x64
    sinkhorn_uv_kernel<<<dim3(nmat), dim3(64), 0, stream>>>(x, out);
}